// GNN_53867479827167
// MI455X (gfx1250) — compile-verified
//
#include <hip/hip_runtime.h>

#define NF_IN  64
#define NF_HID 64
#define NF_OUT 32

typedef __attribute__((ext_vector_type(2))) float v2f;
typedef __attribute__((ext_vector_type(8))) float v8f;

// ---------------------------------------------------------------- utilities
__global__ void fill_kernel(float* __restrict__ p, float val, int n) {
    int i = blockIdx.x * blockDim.x + threadIdx.x;
    if (i < n) p[i] = val;
}

// deg[dst] += 1 per edge (deg pre-initialized to 1.0 for self-loop)
__global__ void count_deg_kernel(const int* __restrict__ dst, float* deg, int E) {
    int e = blockIdx.x * blockDim.x + threadIdx.x;
    if (e < E) atomicAdd(&deg[dst[e]], 1.0f);
}

__global__ void dinv_kernel(float* deg, int n) {
    int i = blockIdx.x * blockDim.x + threadIdx.x;
    if (i < n) deg[i] = rsqrtf(deg[i]);   // deg >= 1 always (self-loop)
}

// ------------------------------------------------- dense GEMM via f32 WMMA
// Y[N x NOUT] = X[N x 64] @ W[64 x NOUT].  One wave per 16x16 output tile,
// K-loop of 16 x V_WMMA_F32_16X16X4_F32.
// A 16x4 layout: lanes 0-15 -> {K=0,K=1}, lanes 16-31 -> {K=2,K=3}, M = lane&15
// B 4x16 layout: mirrored, N = lane&15
// C/D: VGPR j -> M = j (+8 for lanes 16-31), N = lane&15
template<int NOUT>
__global__ void gemm_wmma_kernel(const float* __restrict__ X,
                                 const float* __restrict__ W,
                                 float* __restrict__ Y) {
    const int m0   = blockIdx.x * 16;
    const int n0   = blockIdx.y * 16;
    const int lane = threadIdx.x & 31;
    const int half = lane >> 4;   // 0 or 1
    const int l15  = lane & 15;

    v8f c = {};
    const float* xrow = X + (size_t)(m0 + l15) * 64;
#pragma unroll
    for (int kb = 0; kb < 64; kb += 4) {
        const int ka = kb + half * 2;
        v2f a, b;
        a.x = xrow[ka];
        a.y = xrow[ka + 1];
        b.x = W[(ka + 0) * NOUT + n0 + l15];
        b.y = W[(ka + 1) * NOUT + n0 + l15];
        c = __builtin_amdgcn_wmma_f32_16x16x4_f32(
                /*neg_a=*/false, a, /*neg_b=*/false, b,
                /*c_mod=*/(short)0, c, /*reuse_a=*/false, /*reuse_b=*/false);
    }

    float* yrow = Y + (size_t)(m0 + half * 8) * NOUT + n0 + l15;
#pragma unroll
    for (int j = 0; j < 8; ++j)
        yrow[(size_t)j * NOUT] = c[j];
}

// ------------------------------------------ edge scatter: agg[dst] += norm*h[src]
// One wave per edge; lanes sweep the feature row (coalesced 32-wide).
template<int F>
__global__ void scatter_kernel(const int* __restrict__ src,
                               const int* __restrict__ dst,
                               const float* __restrict__ h,
                               const float* __restrict__ dinv,
                               float* agg, int E) {
    int e    = (blockIdx.x * blockDim.x + threadIdx.x) >> 5;
    int lane = threadIdx.x & 31;
    if (e >= E) return;
    int s = src[e], d = dst[e];
    float nrm = dinv[s] * dinv[d];
    const float* hs = h   + (size_t)s * F;
    float*       ad = agg + (size_t)d * F;
#pragma unroll
    for (int f = lane; f < F; f += 32)
        atomicAdd(&ad[f], hs[f] * nrm);
}

// --------------- add self-loop term (dinv[i]^2 * h_pre) + bias, optional ReLU
template<int F, bool RELU>
__global__ void finalize_kernel(float* __restrict__ agg,
                                const float* __restrict__ hpre,
                                const float* __restrict__ dinv,
                                const float* __restrict__ bias,
                                int total) {
    int idx = blockIdx.x * blockDim.x + threadIdx.x;
    if (idx >= total) return;
    int i = idx / F;
    int f = idx % F;
    float di = dinv[i];
    float v  = agg[idx] + di * di * hpre[idx] + bias[f];
    agg[idx] = RELU ? fmaxf(v, 0.0f) : v;
}

// --------------------------------------------------------------------------
extern "C" void kernel_launch(void* const* d_in, const int* in_sizes, int n_in,
                              void* d_out, int out_size, void* d_ws, size_t ws_size,
                              hipStream_t stream) {
    const float* x  = (const float*)d_in[0];
    const int*   ei = (const int*)  d_in[1];
    const float* W1 = (const float*)d_in[2];
    const float* b1 = (const float*)d_in[3];
    const float* W2 = (const float*)d_in[4];
    const float* b2 = (const float*)d_in[5];

    const int N = in_sizes[0] / NF_IN;   // 50000 (divisible by 16)
    const int E = in_sizes[1] / 2;       // 1000000
    const int* src = ei;
    const int* dst = ei + E;

    // workspace carve-up (~32.2 MB): dinv | h1_pre | agg1(->h1) | h2_pre
    char* ws = (char*)d_ws;
    float* dinv  = (float*)ws;  ws += (((size_t)N * 4) + 255) & ~(size_t)255;
    float* h1pre = (float*)ws;  ws += (size_t)N * NF_HID * 4;
    float* h1    = (float*)ws;  ws += (size_t)N * NF_HID * 4;
    float* h2pre = (float*)ws;
    float* out   = (float*)d_out;

    const int TB = 256;
    auto cdiv = [](int a, int b) { return (a + b - 1) / b; };

    // ---- symmetric normalization: dinv = rsqrt(1 + indegree)
    fill_kernel     <<<cdiv(N, TB), TB, 0, stream>>>(dinv, 1.0f, N);
    count_deg_kernel<<<cdiv(E, TB), TB, 0, stream>>>(dst, dinv, E);
    dinv_kernel     <<<cdiv(N, TB), TB, 0, stream>>>(dinv, N);

    // ---- layer 1: h1 = relu( Dsym (x@W1) + b1 )
    gemm_wmma_kernel<NF_HID><<<dim3(N / 16, NF_HID / 16), 32, 0, stream>>>(x, W1, h1pre);
    fill_kernel<<<cdiv(N * NF_HID, TB), TB, 0, stream>>>(h1, 0.0f, N * NF_HID);
    scatter_kernel<NF_HID><<<cdiv(E * 32, TB), TB, 0, stream>>>(src, dst, h1pre, dinv, h1, E);
    finalize_kernel<NF_HID, true><<<cdiv(N * NF_HID, TB), TB, 0, stream>>>(
        h1, h1pre, dinv, b1, N * NF_HID);

    // ---- layer 2: out = Dsym (h1@W2) + b2
    gemm_wmma_kernel<NF_OUT><<<dim3(N / 16, NF_OUT / 16), 32, 0, stream>>>(h1, W2, h2pre);
    fill_kernel<<<cdiv(N * NF_OUT, TB), TB, 0, stream>>>(out, 0.0f, N * NF_OUT);
    scatter_kernel<NF_OUT><<<cdiv(E * 32, TB), TB, 0, stream>>>(src, dst, h2pre, dinv, out, E);
    finalize_kernel<NF_OUT, false><<<cdiv(N * NF_OUT, TB), TB, 0, stream>>>(
        out, h2pre, dinv, b2, N * NF_OUT);
}